// GraphNeuralNetwork_25013889531943
// MI455X (gfx1250) — compile-verified
//
#include <hip/hip_runtime.h>
#include <math.h>

// GCN 2-layer forward for MI455X (gfx1250, wave32).
// Dense transforms use V_WMMA_F32_16X16X4_F32 (fp32-exact matrix core path).
// Sparse aggregation uses thread-per-edge gather + native global_atomic_add_f32
// (non-returning, relaxed, agent scope) into L2-resident accumulators.

typedef __attribute__((ext_vector_type(2))) float v2f;
typedef __attribute__((ext_vector_type(8))) float v8f;

#define FOUT 16  // HIDDEN == N_CLASSES == 16

// Force the native non-returning f32 atomic add (CDNA5 L2 fadd is denorm-safe).
__device__ __forceinline__ void atomAddF(float* p, float v) {
  (void)__hip_atomic_fetch_add(p, v, __ATOMIC_RELAXED, __HIP_MEMORY_SCOPE_AGENT);
}

// ---------------- degree / normalization ----------------

__global__ void k_zero_u32(unsigned int* __restrict__ p, int n) {
  int i = blockIdx.x * blockDim.x + threadIdx.x;
  if (i < n) p[i] = 0u;
}

__global__ void k_count_deg(const int* __restrict__ dst, unsigned int* __restrict__ cnt, int E) {
  int e = blockIdx.x * blockDim.x + threadIdx.x;
  if (e < E)
    (void)__hip_atomic_fetch_add(&cnt[dst[e]], 1u, __ATOMIC_RELAXED, __HIP_MEMORY_SCOPE_AGENT);
}

__global__ void k_dinv(const unsigned int* __restrict__ cnt, float* __restrict__ dinv, int n) {
  int i = blockIdx.x * blockDim.x + threadIdx.x;
  if (i < n) dinv[i] = rsqrtf((float)cnt[i] + 1.0f);  // +1 for self-loop; deg >= 1 always
}

// ---------------- dense transform: H[M x 16] = A[M x K] @ W[K x 16] ----------------
// One wave per 16-row tile. Chain of K/4 V_WMMA_F32_16X16X4_F32 ops.
// A 16x4: lane L holds row M=L%16; VGPR0/1 = K pair selected by L>>4 (0,1 | 2,3).
// B 4x16 mirrors A: lane holds column N=lane&15, K pair selected by lane>>4.
// D: VGPR r -> M = r + 8*(lane>=16), N = lane&15.
__global__ void __launch_bounds__(256)
k_gemm_wmma(const float* __restrict__ A, const float* __restrict__ W,
            float* __restrict__ H, int M, int K) {
  const int wavesPerBlock = blockDim.x >> 5;
  const int wave = blockIdx.x * wavesPerBlock + (threadIdx.x >> 5);
  const int lane = threadIdx.x & 31;
  const int nTiles = M >> 4;
  if (wave >= nTiles) return;  // wave-uniform: EXEC stays all-1s for active waves

  const int r0    = wave << 4;
  const int mrow  = lane & 15;        // A row / B col / D col
  const int khalf = (lane >> 4) << 1; // 0 or 2

  v8f acc = {};
  const float* __restrict__ arow = A + (size_t)(r0 + mrow) * (size_t)K;

  for (int k = 0; k < K; k += 4) {
    const int kk = k + khalf;
    v2f a = *reinterpret_cast<const v2f*>(arow + kk);  // 8B-aligned (K mult of 4, kk even)
    v2f b;
    b.x = W[(size_t)kk * FOUT + mrow];
    b.y = W[(size_t)(kk + 1) * FOUT + mrow];
    acc = __builtin_amdgcn_wmma_f32_16x16x4_f32(
        /*neg_a=*/false, a, /*neg_b=*/false, b,
        /*c_mod=*/(short)0, acc, /*reuse_a=*/false, /*reuse_b=*/false);
  }

  const int mbase = r0 + ((lane >> 4) << 3);  // +8 rows for upper half-wave
#pragma unroll
  for (int r = 0; r < 8; ++r) {
    H[(size_t)(mbase + r) * FOUT + mrow] = acc[r];
  }
}

// ---------------- sparse aggregation ----------------

// agg[i][:] = dinv[i]^2 * h[i][:]   (self-loop message; also initializes the accumulator)
__global__ void k_init_self(const float* __restrict__ h, const float* __restrict__ dinv,
                            float* __restrict__ agg, int n /* = N*16 */) {
  int idx = blockIdx.x * blockDim.x + threadIdx.x;
  if (idx < n) {
    float di = dinv[idx >> 4];
    agg[idx] = di * di * h[idx];
  }
}

// agg[dst][:] += dinv[src]*dinv[dst] * h[src][:]  for each edge
__global__ void k_edge_agg(const int* __restrict__ src, const int* __restrict__ dst,
                           const float* __restrict__ dinv, const float* __restrict__ h,
                           float* __restrict__ agg, int E) {
  int e = blockIdx.x * blockDim.x + threadIdx.x;
  if (e >= E) return;
  const int s = src[e];
  const int d = dst[e];
  const float norm = dinv[s] * dinv[d];
  const float4* __restrict__ hs = reinterpret_cast<const float4*>(h + (size_t)s * FOUT);
  float* __restrict__ o = agg + (size_t)d * FOUT;
#pragma unroll
  for (int q = 0; q < 4; ++q) {
    float4 v = hs[q];
    atomAddF(o + q * 4 + 0, norm * v.x);
    atomAddF(o + q * 4 + 1, norm * v.y);
    atomAddF(o + q * 4 + 2, norm * v.z);
    atomAddF(o + q * 4 + 3, norm * v.w);
  }
}

// ---------------- epilogues ----------------

__global__ void k_bias_relu(float* __restrict__ a, const float* __restrict__ b, int n) {
  int idx = blockIdx.x * blockDim.x + threadIdx.x;
  if (idx < n) {
    float v = a[idx] + b[idx & (FOUT - 1)];
    a[idx] = v > 0.0f ? v : 0.0f;
  }
}

__global__ void k_bias_sigmoid(float* __restrict__ a, const float* __restrict__ b, int n) {
  int idx = blockIdx.x * blockDim.x + threadIdx.x;
  if (idx < n) {
    float v = a[idx] + b[idx & (FOUT - 1)];
    a[idx] = 1.0f / (1.0f + __expf(-v));
  }
}

// ---------------- launch ----------------

extern "C" void kernel_launch(void* const* d_in, const int* in_sizes, int n_in,
                              void* d_out, int out_size, void* d_ws, size_t ws_size,
                              hipStream_t stream) {
  const float* x  = (const float*)d_in[0];   // [N,128]
  const int*   ei = (const int*)d_in[1];     // [2,E]
  const float* W1 = (const float*)d_in[2];   // [128,16]
  const float* b1 = (const float*)d_in[3];   // [16]
  const float* W2 = (const float*)d_in[4];   // [16,16]
  const float* b2 = (const float*)d_in[5];   // [16]

  const int NF = 128;
  const int N  = in_sizes[0] / NF;   // 100000
  const int E  = in_sizes[1] / 2;    // 6400000
  const int* src = ei;
  const int* dst = ei + E;

  // Carve workspace (256B-aligned slabs), re-initialized every call.
  char* ws = (char*)d_ws;
  size_t off = 0;
  auto carve = [&](size_t bytes) -> char* {
    char* p = ws + off;
    off += (bytes + 255) & ~(size_t)255;
    return p;
  };
  unsigned int* cnt  = (unsigned int*)carve((size_t)N * 4);
  float*        dinv = (float*)carve((size_t)N * 4);
  float*        h1   = (float*)carve((size_t)N * FOUT * 4);
  float*        agg1 = (float*)carve((size_t)N * FOUT * 4);
  float*        h2   = (float*)carve((size_t)N * FOUT * 4);
  float*        outF = (float*)d_out;

  const int B = 256;                    // 8 wave32s per block
  const int wpb = B / 32;
  const int tiles = N / 16;             // 6250 (exact)
  const int gTiles = (tiles + wpb - 1) / wpb;
  const int nElem = N * FOUT;

  // normalization
  k_zero_u32 <<<(N + B - 1) / B, B, 0, stream>>>(cnt, N);
  k_count_deg<<<(E + B - 1) / B, B, 0, stream>>>(dst, cnt, E);
  k_dinv     <<<(N + B - 1) / B, B, 0, stream>>>(cnt, dinv, N);

  // layer 1: h1 = x@W1 ; agg1 = norm-aggregate(h1) ; relu(agg1 + b1)
  k_gemm_wmma  <<<gTiles, B, 0, stream>>>(x, W1, h1, N, NF);
  k_init_self  <<<(nElem + B - 1) / B, B, 0, stream>>>(h1, dinv, agg1, nElem);
  k_edge_agg   <<<(E + B - 1) / B, B, 0, stream>>>(src, dst, dinv, h1, agg1, E);
  k_bias_relu  <<<(nElem + B - 1) / B, B, 0, stream>>>(agg1, b1, nElem);

  // layer 2: h2 = agg1@W2 ; out = norm-aggregate(h2) ; sigmoid(out + b2)
  k_gemm_wmma    <<<gTiles, B, 0, stream>>>(agg1, W2, h2, N, FOUT);
  k_init_self    <<<(nElem + B - 1) / B, B, 0, stream>>>(h2, dinv, outF, nElem);
  k_edge_agg     <<<(E + B - 1) / B, B, 0, stream>>>(src, dst, dinv, h2, outF, E);
  k_bias_sigmoid <<<(nElem + B - 1) / B, B, 0, stream>>>(outF, b2, nElem);
}